// MultiHeadSelfAttn_86028194939536
// MI455X (gfx1250) — compile-verified
//
#include <hip/hip_runtime.h>

#define N_HEAD 8
#define D_HEAD 64
#define D_MODEL 512
#define KLEN 8192
#define BSZ 32
#define SCALE 0.125f   // 1/sqrt(64)
#define CLIP 10.0f

typedef __attribute__((ext_vector_type(2))) float v2f;
typedef __attribute__((ext_vector_type(8))) float v8f;

// One block per (batch, head). 256 threads = 8 wave32.
// Each wave streams 1024 K/V rows in 16-row tiles:
//   scores: v_wmma_f32_16x16x4_f32 (K-tile x q-broadcast, chained over d)
//   w = exp(10*tanh(s/8)); online (max-free) softmax accumulation:
//   num[64] += w * V-row (coalesced float2/lane), den += w.
__global__ __launch_bounds__(256)
void mhsa_decode_kernel(const float* __restrict__ q,
                        const float* __restrict__ k,
                        const float* __restrict__ v,
                        float* __restrict__ out) {
    const int bh   = blockIdx.x;        // 0..255
    const int b    = bh >> 3;
    const int n    = bh & 7;
    const int lane = threadIdx.x & 31;
    const int wave = threadIdx.x >> 5;
    const int half = lane >> 4;         // 0: K-pair {0,1} / rows 0-7, 1: {2,3} / rows 8-15
    const int m    = lane & 15;         // A-matrix row within tile

    const size_t base  = (size_t)b * D_MODEL + (size_t)n * D_HEAD;
    const float* qp = q + base;
    const float* kp = k + base;         // row stride D_MODEL*BSZ floats
    const float* vp = v + base;
    const size_t rstride = (size_t)D_MODEL * BSZ;   // 16384 floats = 64 KB

    // Preload q into WMMA B layout (replicated across all 16 columns):
    // lane holds B[2*half + {0,1}, *] = q[4c + 2*half + {0,1}] for chunk c.
    v2f bq[16];
#pragma unroll
    for (int c = 0; c < 16; ++c)
        bq[c] = *(const v2f*)(qp + 4 * c + 2 * half);

    v2f  acc = {0.0f, 0.0f};            // num for d = 2*lane, 2*lane+1
    float den = 0.0f;

    for (int t = 0; t < 64; ++t) {
        const int j0 = wave * 1024 + t * 16;

        // ---- A tile: K rows in WMMA 16x4-f32 A layout (8B-aligned loads) ----
        const float* krow = kp + (size_t)(j0 + m) * rstride;
        v2f a[16];
#pragma unroll
        for (int c = 0; c < 16; ++c)
            a[c] = *(const v2f*)(krow + 4 * c + 2 * half);

        // Prefetch next tile's K/V rows (speculative: tail OOB is dropped).
        if (t + 1 < 64) {
            const size_t jn = (size_t)(j0 + 16 + m) * rstride + 32 * half;
            __builtin_prefetch(kp + jn, 0, 1);
            __builtin_prefetch(vp + jn, 0, 1);
        }

        // ---- scores: two independent f32 WMMA accumulation chains over d ----
        v8f c0 = {}, c1 = {};
#pragma unroll
        for (int c = 0; c < 16; c += 2) {
            c0 = __builtin_amdgcn_wmma_f32_16x16x4_f32(
                     false, a[c],     false, bq[c],     (short)0, c0, false, false);
            c1 = __builtin_amdgcn_wmma_f32_16x16x4_f32(
                     false, a[c + 1], false, bq[c + 1], (short)0, c1, false, false);
        }

        // ---- w = exp(10 * tanh(s * SCALE)); rows (0-7 | 8-15) per half ----
        float w8[8];
#pragma unroll
        for (int r = 0; r < 8; ++r) {
            float s = (c0[r] + c1[r]) * SCALE;
            s = fminf(fmaxf(s, -15.0f), 15.0f);          // tanh saturated anyway
            float e2 = __expf(2.0f * s);
            float th = (e2 - 1.0f) * __builtin_amdgcn_rcpf(e2 + 1.0f);
            w8[r] = __expf(CLIP * th);
        }

        // ---- A·V: lane owns d = {2*lane, 2*lane+1}; coalesced 256B rows ----
        const float* vrow = vp + (size_t)j0 * rstride + 2 * lane;
#pragma unroll
        for (int i = 0; i < 16; ++i) {
            const float wj = __shfl(w8[i & 7], (i < 8) ? 0 : 16, 32);
            const v2f vv = *(const v2f*)(vrow + (size_t)i * rstride);
            acc.x += wj * vv.x;
            acc.y += wj * vv.y;
            den   += wj;                  // identical across lanes
        }
    }

    // ---- cross-wave reduction ----
    __shared__ float s_num[8][64];
    __shared__ float s_den[8];
    s_num[wave][2 * lane]     = acc.x;
    s_num[wave][2 * lane + 1] = acc.y;
    if (lane == 0) s_den[wave] = den;
    __syncthreads();

    if (threadIdx.x < 64) {
        float num = 0.0f, dn = 0.0f;
#pragma unroll
        for (int w = 0; w < 8; ++w) {
            num += s_num[w][threadIdx.x];
            dn  += s_den[w];
        }
        out[base + threadIdx.x] = num / dn;
    }
}

extern "C" void kernel_launch(void* const* d_in, const int* in_sizes, int n_in,
                              void* d_out, int out_size, void* d_ws, size_t ws_size,
                              hipStream_t stream) {
    (void)in_sizes; (void)n_in; (void)d_ws; (void)ws_size; (void)out_size;
    const float* q = (const float*)d_in[0];
    const float* k = (const float*)d_in[1];
    const float* v = (const float*)d_in[2];
    float* out = (float*)d_out;
    mhsa_decode_kernel<<<dim3(BSZ * N_HEAD), dim3(256), 0, stream>>>(q, k, v, out);
}